// GCNEncoder_36550171689598
// MI455X (gfx1250) — compile-verified
//
#include <hip/hip_runtime.h>

typedef float v2f __attribute__((ext_vector_type(2)));
typedef float v8f __attribute__((ext_vector_type(8)));
typedef float v4f __attribute__((ext_vector_type(4)));

// ---------------------------------------------------------------------------
// Degree / normalization
// ---------------------------------------------------------------------------
__global__ void k_init_deg(float* deg, int n) {
    int i = blockIdx.x * blockDim.x + threadIdx.x;
    if (i < n) deg[i] = 1.0f;                 // self-loop contributes 1
}

__global__ void k_edge_deg(const long long* __restrict__ dst, int ne, float* deg) {
    int e = blockIdx.x * blockDim.x + threadIdx.x;
    if (e < ne) atomicAdd(&deg[(int)dst[e]], 1.0f);
}

__global__ void k_finish_dis(float* deg, int n) {
    int i = blockIdx.x * blockDim.x + threadIdx.x;
    if (i < n) deg[i] = rsqrtf(deg[i]);       // deg >= 1 always (self-loop)
}

// ---------------------------------------------------------------------------
// GEMM: H[N x F] = A[N x K] * W[K x F], fp32 WMMA 16x16x4.
// Epilogue also writes AGG = dis[row]^2 * H  (self-loop term of aggregation).
// One wave32 per 16x16 output tile; EXEC all ones as WMMA requires.
// ---------------------------------------------------------------------------
__global__ __launch_bounds__(32)
void k_gemm_wmma(const float* __restrict__ A, const float* __restrict__ W,
                 const float* __restrict__ dis,
                 float* __restrict__ H, float* __restrict__ AGG,
                 int K, int F) {
    const int row0 = blockIdx.x * 16;
    const int col0 = blockIdx.y * 16;
    const int lane = threadIdx.x;
    const int half = lane >> 4;     // 0: lanes 0-15, 1: lanes 16-31
    const int l    = lane & 15;

    const float* arow = A + (size_t)(row0 + l) * K;

    v8f acc = {};
#if __has_builtin(__builtin_amdgcn_wmma_f32_16x16x4_f32)
    for (int k0 = 0; k0 < K; k0 += 4) {
        // A-frag (16x4 f32): lanes 0-15 hold K=k0,k0+1; lanes 16-31 K=k0+2,k0+3
        v2f a = *(const v2f*)(arow + k0 + 2 * half);
        // B-frag (4x16 f32): VGPR0 = row k0 (lanes 0-15) / k0+2 (16-31); VGPR1 = +1
        const float* wp = W + (size_t)(k0 + 2 * half) * F + col0 + l;
        v2f b;
        b.x = wp[0];
        b.y = wp[F];
        acc = __builtin_amdgcn_wmma_f32_16x16x4_f32(
            /*neg_a=*/false, a, /*neg_b=*/false, b,
            /*c_mod=*/(short)0, acc, /*reuse_a=*/false, /*reuse_b=*/false);
    }
    // C/D layout: VGPR v -> M = v (lanes 0-15) or v+8 (lanes 16-31), N = l
    #pragma unroll
    for (int v = 0; v < 8; ++v) {
        int row = row0 + v + 8 * half;
        float val = acc[v];
        size_t idx = (size_t)row * F + col0 + l;
        float d = dis[row];
        H[idx]   = val;
        AGG[idx] = d * d * val;
    }
#else
    // scalar fallback (should not be taken on gfx1250)
    for (int m = 0; m < 16; ++m) {
        if (lane < 16) {
            float s = 0.f;
            for (int k = 0; k < K; ++k) s += A[(size_t)(row0 + m) * K + k] * W[(size_t)k * F + col0 + l];
            size_t idx = (size_t)(row0 + m) * F + col0 + l;
            float d = dis[row0 + m];
            H[idx] = s;
            AGG[idx] = d * d * s;
        }
    }
#endif
}

// ---------------------------------------------------------------------------
// Edge aggregation: AGG[dst] += dis[src]*dis[dst] * H[src]
// One wave32 per edge; each lane handles FPL contiguous features.
// F = FPL*32 (FPL=4 -> F=128, FPL=2 -> F=64). Gathers hit L2 (H is 25.6MB).
// ---------------------------------------------------------------------------
template <int FPL>
__global__ void k_edge_agg(const long long* __restrict__ srcs,
                           const long long* __restrict__ dsts, int ne,
                           const float* __restrict__ dis,
                           const float* __restrict__ H,
                           float* AGG) {
    int gid  = blockIdx.x * blockDim.x + threadIdx.x;
    int e    = gid >> 5;
    int lane = gid & 31;
    if (e >= ne) return;
    int s = (int)srcs[e];
    int d = (int)dsts[e];
    float norm = dis[s] * dis[d];
    const float* hrow = H   + (size_t)s * (FPL * 32) + lane * FPL;
    float*       arow = AGG + (size_t)d * (FPL * 32) + lane * FPL;
    float v[FPL];
    #pragma unroll
    for (int j = 0; j < FPL; ++j) v[j] = hrow[j];       // contiguous -> b128/b64 load
    #pragma unroll
    for (int j = 0; j < FPL; ++j) atomicAdd(&arow[j], norm * v[j]);
}

// ---------------------------------------------------------------------------
// Bias (+ optional ReLU): OUT[i] = act(AGG[i] + b[i & fmask])
// ---------------------------------------------------------------------------
__global__ void k_bias_act(const float* __restrict__ AGG, const float* __restrict__ bias,
                           float* __restrict__ out, int n, int fmask, int relu) {
    int i = blockIdx.x * blockDim.x + threadIdx.x;
    if (i < n) {
        float v = AGG[i] + bias[i & fmask];
        out[i] = relu ? fmaxf(v, 0.0f) : v;
    }
}

// ---------------------------------------------------------------------------
extern "C" void kernel_launch(void* const* d_in, const int* in_sizes, int n_in,
                              void* d_out, int out_size, void* d_ws, size_t ws_size,
                              hipStream_t stream) {
    const float*     x   = (const float*)d_in[0];
    const long long* ei  = (const long long*)d_in[1];   // int64 edge_index [2, E]
    const float*     W1  = (const float*)d_in[2];
    const float*     b1  = (const float*)d_in[3];
    const float*     W2  = (const float*)d_in[4];
    const float*     b2  = (const float*)d_in[5];
    const float*     W3  = (const float*)d_in[6];
    const float*     b3  = (const float*)d_in[7];

    const int IN_C = 128, HID = 128, OUT_C = 64;
    const int N  = in_sizes[0] / IN_C;      // 50000
    const int NE = in_sizes[1] / 2;         // 800000
    const long long* src = ei;
    const long long* dst = ei + NE;

    // Workspace: dis[N] | bufH[N*128] | bufA[N*128] | bufB[N*128]
    float* ws   = (float*)d_ws;
    size_t nPad = ((size_t)N + 63) & ~(size_t)63;
    float* dis  = ws;
    float* bufH = ws + nPad;
    float* bufA = bufH + (size_t)N * HID;
    float* bufB = bufA + (size_t)N * HID;

    const int TPB = 256;
    dim3 blk(TPB);

    // --- normalization coefficients ---
    k_init_deg  <<<(N  + TPB - 1) / TPB, blk, 0, stream>>>(dis, N);
    k_edge_deg  <<<(NE + TPB - 1) / TPB, blk, 0, stream>>>(dst, NE, dis);
    k_finish_dis<<<(N  + TPB - 1) / TPB, blk, 0, stream>>>(dis, N);

    const int eblocks = (NE * 32 + TPB - 1) / TPB;

    // --- layer 1: x[N,128] @ W1[128,128] ---
    k_gemm_wmma<<<dim3(N / 16, HID / 16), dim3(32), 0, stream>>>(x, W1, dis, bufH, bufA, IN_C, HID);
    k_edge_agg<4><<<eblocks, blk, 0, stream>>>(src, dst, NE, dis, bufH, bufA);
    k_bias_act<<<((N * HID) + TPB - 1) / TPB, blk, 0, stream>>>(bufA, b1, bufB, N * HID, HID - 1, 1);

    // --- layer 2: a1[N,128] @ W2[128,128] ---
    k_gemm_wmma<<<dim3(N / 16, HID / 16), dim3(32), 0, stream>>>(bufB, W2, dis, bufH, bufA, HID, HID);
    k_edge_agg<4><<<eblocks, blk, 0, stream>>>(src, dst, NE, dis, bufH, bufA);
    k_bias_act<<<((N * HID) + TPB - 1) / TPB, blk, 0, stream>>>(bufA, b2, bufB, N * HID, HID - 1, 1);

    // --- layer 3: a2[N,128] @ W3[128,64] (no ReLU) ---
    k_gemm_wmma<<<dim3(N / 16, OUT_C / 16), dim3(32), 0, stream>>>(bufB, W3, dis, bufH, bufA, HID, OUT_C);
    k_edge_agg<2><<<eblocks, blk, 0, stream>>>(src, dst, NE, dis, bufH, bufA);
    k_bias_act<<<((N * OUT_C) + TPB - 1) / TPB, blk, 0, stream>>>(bufA, b3, (float*)d_out, N * OUT_C, OUT_C - 1, 0);
}